// LerayProjector_88441966559593
// MI455X (gfx1250) — compile-verified
//
#include <hip/hip_runtime.h>
#include <hip/hip_bf16.h>

// ---------------------------------------------------------------------------
// Leray projection on a graph (MI455X / gfx1250, wave32).
//   rhs = div(u);  CG solve L p = rhs (zero-mean);  out = u - grad(p)
//
// Everything (node fields, u_raw, indices) fits in the 192MB L2, so the edge
// gather/scatter kernels are L2-bandwidth / atomic-throughput bound. Node
// fields are stored node-major [N][8] so each edge endpoint is one contiguous
// 32B load and one contiguous 64B atomic region. Scatter-adds use Q32.32
// fixed point with u64 integer atomics => bit-deterministic results.
// ---------------------------------------------------------------------------

#define NND 50000
#define NED 1600000
#define NB 8
#define MAXIT 50
#define TOL2 1.0e-8f      // TOL*TOL
#define EPSF 1.0e-12f

#define EDGE_BLKS 6250    // NED / 256
#define NODE_BLKS 196     // ceil(NND / 256)
#define NBLK_RED 128      // fixed grid for reduction kernels (partials count)
#define RED_BS 256

#define SCALE_F   4294967296.0f            // 2^32
#define INV_SCALE 2.3283064365386963e-10   // 2^-32

// scal[] float slots
#define SC_ALPHA   0
#define SC_MEANAP  8
#define SC_BETA    16
#define SC_MEANR   24
#define SC_RSOLD   32
#define SC_RHSN    40
#define SC_MEANRHS 48
#define SC_ACTIVE  56
#define SC_COUNT   64

__device__ __forceinline__ long long f2fx(float g) {
  return (long long)rintf(g * SCALE_F);
}
__device__ __forceinline__ float fx2f(long long v) {
  return (float)((double)v * INV_SCALE);
}

template <int BS>
__device__ __forceinline__ float blk_sum(float v, float* sbuf) {
  int t = threadIdx.x;
  sbuf[t] = v;
  __syncthreads();
#pragma unroll
  for (int s = BS / 2; s > 0; s >>= 1) {
    if (t < s) sbuf[t] += sbuf[t + s];
    __syncthreads();
  }
  float r = sbuf[0];
  __syncthreads();
  return r;
}

// ------------------------------ setup --------------------------------------

__global__ __launch_bounds__(256) void k_zero(long long* apf) {
  for (int i = blockIdx.x * blockDim.x + threadIdx.x; i < NND * NB;
       i += gridDim.x * blockDim.x)
    apf[i] = 0;
}

// rhs = div(u): scatter +u at src, -u at dst (fixed-point atomics).
__global__ __launch_bounds__(256) void k_div_scatter(const float* __restrict__ u,
                                                     const int* __restrict__ src,
                                                     const int* __restrict__ dst,
                                                     long long* __restrict__ apf) {
  for (int e = blockIdx.x * blockDim.x + threadIdx.x; e < NED;
       e += gridDim.x * blockDim.x) {
    if (e + 8192 < NED) {
      __builtin_prefetch(&src[e + 8192], 0, 1);
      __builtin_prefetch(&dst[e + 8192], 0, 1);
      __builtin_prefetch(&u[e + 8192], 0, 1);
    }
    int s = src[e], d = dst[e];
    unsigned long long* As = (unsigned long long*)(apf + (size_t)s * NB);
    unsigned long long* Ad = (unsigned long long*)(apf + (size_t)d * NB);
#pragma unroll
    for (int b = 0; b < NB; b++) {
      long long q = f2fx(u[b * NED + e]);
      atomicAdd(As + b, (unsigned long long)q);
      atomicAdd(Ad + b, (unsigned long long)(-q));
    }
  }
}

// partials of sum(rhs), sum(rhs^2) per batch  -> part2[q*NBLK_RED + blk]
__global__ __launch_bounds__(RED_BS) void k_setup_reduce(const long long* __restrict__ apf,
                                                         float* __restrict__ part2) {
  __shared__ float sbuf[RED_BS];
  float accS[NB], accQ[NB];
#pragma unroll
  for (int b = 0; b < NB; b++) { accS[b] = 0.f; accQ[b] = 0.f; }
  for (int n = blockIdx.x * blockDim.x + threadIdx.x; n < NND;
       n += gridDim.x * blockDim.x) {
#pragma unroll
    for (int b = 0; b < NB; b++) {
      float v = fx2f(apf[n * NB + b]);
      accS[b] += v;
      accQ[b] += v * v;
    }
  }
#pragma unroll
  for (int b = 0; b < NB; b++) {
    float s = blk_sum<RED_BS>(accS[b], sbuf);
    if (threadIdx.x == 0) part2[b * NBLK_RED + blockIdx.x] = s;
  }
#pragma unroll
  for (int b = 0; b < NB; b++) {
    float s = blk_sum<RED_BS>(accQ[b], sbuf);
    if (threadIdx.x == 0) part2[(NB + b) * NBLK_RED + blockIdx.x] = s;
  }
}

__global__ __launch_bounds__(NBLK_RED) void k_setup_scalars(const float* __restrict__ part2,
                                                            float* __restrict__ scal,
                                                            int* __restrict__ done) {
  __shared__ float sbuf[NBLK_RED];
  __shared__ float tot[2 * NB];
  int t = threadIdx.x;
  for (int q = 0; q < 2 * NB; q++) {
    float s = blk_sum<NBLK_RED>(part2[q * NBLK_RED + t], sbuf);
    if (t == 0) tot[q] = s;
  }
  __syncthreads();
  if (t < NB) {
    float mean = tot[t] * (1.0f / NND);
    float nrm = tot[NB + t] - (float)NND * mean * mean;  // sum((x-m)^2)
    if (nrm < 0.f) nrm = 0.f;
    scal[SC_MEANRHS + t] = mean;
    scal[SC_RHSN + t] = nrm;
    scal[SC_RSOLD + t] = nrm;
  }
  if (t == 0) *done = 0;
}

// r = p = zero_mean(rhs), x = 0, apf cleared for first Laplacian apply.
__global__ __launch_bounds__(256) void k_setup_nodes(long long* __restrict__ apf,
                                                     const float* __restrict__ scal,
                                                     float* __restrict__ x,
                                                     float* __restrict__ r,
                                                     float* __restrict__ p) {
  float mean[NB];
#pragma unroll
  for (int b = 0; b < NB; b++) mean[b] = scal[SC_MEANRHS + b];
  for (int n = blockIdx.x * blockDim.x + threadIdx.x; n < NND;
       n += gridDim.x * blockDim.x) {
#pragma unroll
    for (int b = 0; b < NB; b++) {
      int i = n * NB + b;
      float v = fx2f(apf[i]) - mean[b];
      r[i] = v;
      p[i] = v;
      x[i] = 0.f;
      apf[i] = 0;
    }
  }
}

// ---------------------------- CG iteration ---------------------------------

// Ap_fixed += L p  (edge-parallel fused gradient+divergence, u64 atomics)
__global__ __launch_bounds__(256) void k_lap_edges(const int* __restrict__ src,
                                                   const int* __restrict__ dst,
                                                   const float* __restrict__ p,
                                                   long long* __restrict__ apf,
                                                   const int* __restrict__ done) {
  if (*done) return;
  for (int e = blockIdx.x * blockDim.x + threadIdx.x; e < NED;
       e += gridDim.x * blockDim.x) {
    if (e + 8192 < NED) {
      __builtin_prefetch(&src[e + 8192], 0, 1);
      __builtin_prefetch(&dst[e + 8192], 0, 1);
    }
    int s = src[e], d = dst[e];
    const float4* ps = (const float4*)(p + (size_t)s * NB);
    const float4* pd = (const float4*)(p + (size_t)d * NB);
    float4 a0 = ps[0], a1 = ps[1];
    float4 c0 = pd[0], c1 = pd[1];
    unsigned long long* As = (unsigned long long*)(apf + (size_t)s * NB);
    unsigned long long* Ad = (unsigned long long*)(apf + (size_t)d * NB);
    long long q;
    q = f2fx(a0.x - c0.x); atomicAdd(As + 0, (unsigned long long)q); atomicAdd(Ad + 0, (unsigned long long)(-q));
    q = f2fx(a0.y - c0.y); atomicAdd(As + 1, (unsigned long long)q); atomicAdd(Ad + 1, (unsigned long long)(-q));
    q = f2fx(a0.z - c0.z); atomicAdd(As + 2, (unsigned long long)q); atomicAdd(Ad + 2, (unsigned long long)(-q));
    q = f2fx(a0.w - c0.w); atomicAdd(As + 3, (unsigned long long)q); atomicAdd(Ad + 3, (unsigned long long)(-q));
    q = f2fx(a1.x - c1.x); atomicAdd(As + 4, (unsigned long long)q); atomicAdd(Ad + 4, (unsigned long long)(-q));
    q = f2fx(a1.y - c1.y); atomicAdd(As + 5, (unsigned long long)q); atomicAdd(Ad + 5, (unsigned long long)(-q));
    q = f2fx(a1.z - c1.z); atomicAdd(As + 6, (unsigned long long)q); atomicAdd(Ad + 6, (unsigned long long)(-q));
    q = f2fx(a1.w - c1.w); atomicAdd(As + 7, (unsigned long long)q); atomicAdd(Ad + 7, (unsigned long long)(-q));
  }
}

// partials of sum(Ap), sum(p*Ap), sum(p) per batch -> part1
__global__ __launch_bounds__(RED_BS) void k_reduce1(const long long* __restrict__ apf,
                                                    const float* __restrict__ p,
                                                    float* __restrict__ part1,
                                                    const int* __restrict__ done) {
  if (*done) return;
  __shared__ float sbuf[RED_BS];
  float aS[NB], aD[NB], aP[NB];
#pragma unroll
  for (int b = 0; b < NB; b++) { aS[b] = 0.f; aD[b] = 0.f; aP[b] = 0.f; }
  for (int n = blockIdx.x * blockDim.x + threadIdx.x; n < NND;
       n += gridDim.x * blockDim.x) {
#pragma unroll
    for (int b = 0; b < NB; b++) {
      float ap = fx2f(apf[n * NB + b]);
      float pv = p[n * NB + b];
      aS[b] += ap;
      aD[b] += pv * ap;
      aP[b] += pv;
    }
  }
#pragma unroll
  for (int b = 0; b < NB; b++) {
    float s = blk_sum<RED_BS>(aS[b], sbuf);
    if (threadIdx.x == 0) part1[b * NBLK_RED + blockIdx.x] = s;
  }
#pragma unroll
  for (int b = 0; b < NB; b++) {
    float s = blk_sum<RED_BS>(aD[b], sbuf);
    if (threadIdx.x == 0) part1[(NB + b) * NBLK_RED + blockIdx.x] = s;
  }
#pragma unroll
  for (int b = 0; b < NB; b++) {
    float s = blk_sum<RED_BS>(aP[b], sbuf);
    if (threadIdx.x == 0) part1[(2 * NB + b) * NBLK_RED + blockIdx.x] = s;
  }
}

// alpha = active ? rs_old / (p . zero_mean(Ap)) : 0
__global__ __launch_bounds__(NBLK_RED) void k_scalars1(const float* __restrict__ part1,
                                                       float* __restrict__ scal,
                                                       const int* __restrict__ done) {
  if (*done) return;
  __shared__ float sbuf[NBLK_RED];
  __shared__ float tot[3 * NB];
  int t = threadIdx.x;
  for (int q = 0; q < 3 * NB; q++) {
    float s = blk_sum<NBLK_RED>(part1[q * NBLK_RED + t], sbuf);
    if (t == 0) tot[q] = s;
  }
  __syncthreads();
  if (t < NB) {
    float sumAp = tot[t], dot = tot[NB + t], sump = tot[2 * NB + t];
    float meanAp = sumAp * (1.0f / NND);
    float denom = dot - meanAp * sump;  // == p . (Ap - meanAp)
    float rs_old = scal[SC_RSOLD + t];
    float active = (rs_old > EPSF && fabsf(denom) > EPSF) ? 1.0f : 0.0f;
    float alpha = (active != 0.0f) ? rs_old / denom : 0.0f;
    scal[SC_ALPHA + t] = alpha;
    scal[SC_MEANAP + t] = meanAp;
    scal[SC_ACTIVE + t] = active;
  }
}

// x += alpha p ; r -= alpha*(Ap-meanAp) ; clear Ap_fixed ; partials of r, r^2
__global__ __launch_bounds__(RED_BS) void k_update_xr(long long* __restrict__ apf,
                                                      const float* __restrict__ scal,
                                                      float* __restrict__ x,
                                                      float* __restrict__ r,
                                                      const float* __restrict__ p,
                                                      float* __restrict__ part2,
                                                      const int* __restrict__ done) {
  if (*done) return;
  __shared__ float sbuf[RED_BS];
  float alpha[NB], meanAp[NB], aS[NB], aQ[NB];
#pragma unroll
  for (int b = 0; b < NB; b++) {
    alpha[b] = scal[SC_ALPHA + b];
    meanAp[b] = scal[SC_MEANAP + b];
    aS[b] = 0.f;
    aQ[b] = 0.f;
  }
  for (int n = blockIdx.x * blockDim.x + threadIdx.x; n < NND;
       n += gridDim.x * blockDim.x) {
#pragma unroll
    for (int b = 0; b < NB; b++) {
      int i = n * NB + b;
      float ap = fx2f(apf[i]) - meanAp[b];
      apf[i] = 0;
      x[i] += alpha[b] * p[i];
      float rn = r[i] - alpha[b] * ap;
      r[i] = rn;
      aS[b] += rn;
      aQ[b] += rn * rn;
    }
  }
#pragma unroll
  for (int b = 0; b < NB; b++) {
    float s = blk_sum<RED_BS>(aS[b], sbuf);
    if (threadIdx.x == 0) part2[b * NBLK_RED + blockIdx.x] = s;
  }
#pragma unroll
  for (int b = 0; b < NB; b++) {
    float s = blk_sum<RED_BS>(aQ[b], sbuf);
    if (threadIdx.x == 0) part2[(NB + b) * NBLK_RED + blockIdx.x] = s;
  }
}

// rs_new, beta, convergence flag
__global__ __launch_bounds__(NBLK_RED) void k_scalars2(const float* __restrict__ part2,
                                                       float* __restrict__ scal,
                                                       int* __restrict__ done) {
  if (*done) return;
  __shared__ float sbuf[NBLK_RED];
  __shared__ float tot[2 * NB];
  int t = threadIdx.x;
  for (int q = 0; q < 2 * NB; q++) {
    float s = blk_sum<NBLK_RED>(part2[q * NBLK_RED + t], sbuf);
    if (t == 0) tot[q] = s;
  }
  __syncthreads();
  if (t == 0) {
    float maxrs = 0.f, maxrel = 0.f;
    for (int b = 0; b < NB; b++) {
      float mean_r = tot[b] * (1.0f / NND);
      float rs_new = tot[NB + b] - (float)NND * mean_r * mean_r;  // sum((r-m)^2)
      if (rs_new < 0.f) rs_new = 0.f;
      float rel = rs_new / fmaxf(scal[SC_RHSN + b], EPSF);
      maxrs = fmaxf(maxrs, rs_new);
      maxrel = fmaxf(maxrel, rel);
      float beta = (scal[SC_ACTIVE + b] != 0.0f) ? rs_new / scal[SC_RSOLD + b] : 0.0f;
      scal[SC_BETA + b] = beta;
      scal[SC_MEANR + b] = mean_r;
      scal[SC_RSOLD + b] = rs_new;
    }
    if (maxrs < TOL2 && maxrel < TOL2) *done = 1;
  }
}

// r = zero_mean(r) ; p = r + beta p
__global__ __launch_bounds__(256) void k_update_p(const float* __restrict__ scal,
                                                  float* __restrict__ r,
                                                  float* __restrict__ p,
                                                  const int* __restrict__ done) {
  if (*done) return;
  float beta[NB], mean_r[NB];
#pragma unroll
  for (int b = 0; b < NB; b++) {
    beta[b] = scal[SC_BETA + b];
    mean_r[b] = scal[SC_MEANR + b];
  }
  for (int n = blockIdx.x * blockDim.x + threadIdx.x; n < NND;
       n += gridDim.x * blockDim.x) {
#pragma unroll
    for (int b = 0; b < NB; b++) {
      int i = n * NB + b;
      float rn = r[i] - mean_r[b];
      r[i] = rn;
      p[i] = rn + beta[b] * p[i];
    }
  }
}

// ------------------------------- final -------------------------------------

// out = u - grad(x)   (grad of the zero-mean shift cancels, so skip it)
__global__ __launch_bounds__(256) void k_project(const float* __restrict__ u,
                                                 const int* __restrict__ src,
                                                 const int* __restrict__ dst,
                                                 const float* __restrict__ x,
                                                 float* __restrict__ out) {
  for (int e = blockIdx.x * blockDim.x + threadIdx.x; e < NED;
       e += gridDim.x * blockDim.x) {
    if (e + 8192 < NED) {
      __builtin_prefetch(&src[e + 8192], 0, 1);
      __builtin_prefetch(&dst[e + 8192], 0, 1);
    }
    int s = src[e], d = dst[e];
    const float4* xs = (const float4*)(x + (size_t)s * NB);
    const float4* xd = (const float4*)(x + (size_t)d * NB);
    float4 a0 = xs[0], a1 = xs[1];
    float4 c0 = xd[0], c1 = xd[1];
    out[0 * NED + e] = u[0 * NED + e] - (a0.x - c0.x);
    out[1 * NED + e] = u[1 * NED + e] - (a0.y - c0.y);
    out[2 * NED + e] = u[2 * NED + e] - (a0.z - c0.z);
    out[3 * NED + e] = u[3 * NED + e] - (a0.w - c0.w);
    out[4 * NED + e] = u[4 * NED + e] - (a1.x - c1.x);
    out[5 * NED + e] = u[5 * NED + e] - (a1.y - c1.y);
    out[6 * NED + e] = u[6 * NED + e] - (a1.z - c1.z);
    out[7 * NED + e] = u[7 * NED + e] - (a1.w - c1.w);
  }
}

// ---------------------------------------------------------------------------

extern "C" void kernel_launch(void* const* d_in, const int* in_sizes, int n_in,
                              void* d_out, int out_size, void* d_ws, size_t ws_size,
                              hipStream_t stream) {
  const float* u = (const float*)d_in[0];
  const int* src = (const int*)d_in[1];
  const int* dst = (const int*)d_in[2];
  float* out = (float*)d_out;

  char* ws = (char*)d_ws;
  auto au = [](size_t v) { return (v + 255) & ~(size_t)255; };
  size_t o = 0;
  long long* apf = (long long*)(ws + o); o += au((size_t)NND * NB * sizeof(long long));
  float* x = (float*)(ws + o);     o += au((size_t)NND * NB * sizeof(float));
  float* r = (float*)(ws + o);     o += au((size_t)NND * NB * sizeof(float));
  float* p = (float*)(ws + o);     o += au((size_t)NND * NB * sizeof(float));
  float* part1 = (float*)(ws + o); o += au((size_t)3 * NB * NBLK_RED * sizeof(float));
  float* part2 = (float*)(ws + o); o += au((size_t)2 * NB * NBLK_RED * sizeof(float));
  float* scal = (float*)(ws + o);  o += au(SC_COUNT * sizeof(float));
  int* done = (int*)(ws + o);      o += au(sizeof(int));
  (void)ws_size; (void)in_sizes; (void)n_in; (void)out_size;

  // setup: rhs = zero_mean(div(u)); r = p = rhs; x = 0
  k_zero<<<1563, 256, 0, stream>>>(apf);
  k_div_scatter<<<EDGE_BLKS, 256, 0, stream>>>(u, src, dst, apf);
  k_setup_reduce<<<NBLK_RED, RED_BS, 0, stream>>>(apf, part2);
  k_setup_scalars<<<1, NBLK_RED, 0, stream>>>(part2, scal, done);
  k_setup_nodes<<<NODE_BLKS, 256, 0, stream>>>(apf, scal, x, r, p);

  // CG iterations; device-side `done` flag implements early termination
  // under graph capture (converged iterations become near-empty launches).
  for (int it = 0; it < MAXIT; ++it) {
    k_lap_edges<<<EDGE_BLKS, 256, 0, stream>>>(src, dst, p, apf, done);
    k_reduce1<<<NBLK_RED, RED_BS, 0, stream>>>(apf, p, part1, done);
    k_scalars1<<<1, NBLK_RED, 0, stream>>>(part1, scal, done);
    k_update_xr<<<NBLK_RED, RED_BS, 0, stream>>>(apf, scal, x, r, p, part2, done);
    k_scalars2<<<1, NBLK_RED, 0, stream>>>(part2, scal, done);
    k_update_p<<<NODE_BLKS, 256, 0, stream>>>(scal, r, p, done);
  }

  // out = u - grad(x)
  k_project<<<EDGE_BLKS, 256, 0, stream>>>(u, src, dst, x, out);
}